// MPNNSingleHAttn_34643206210179
// MI455X (gfx1250) — compile-verified
//
#include <hip/hip_runtime.h>
#include <hip/hip_bf16.h>
#include <math.h>

#define BB 4
#define NN 1024
#define HH 64
#define DD 64
#define FEF 16

typedef float v2f __attribute__((ext_vector_type(2)));
typedef float v8f __attribute__((ext_vector_type(8)));

// ---------------------------------------------------------------------------
// Kernel 1: z[b,n,d] = sum_h h0[b,n,h] * w_fc[d,h]
// One wave (32 threads) computes a 16x16 tile of z using V_WMMA_F32_16X16X4_F32.
// A (16x4, f32): lanes 0-15 hold K=k0,k0+1 in v0,v1; lanes 16-31 hold K=k0+2,k0+3.
// B (4x16, f32): row k striped across lanes: lanes 0-15 = K=k0/k0+1, lanes 16-31 = K=k0+2/k0+3.
// C/D (16x16, f32): VGPR v holds M = v + 8*(lane>=16), N = lane&15.
// ---------------------------------------------------------------------------
__global__ __launch_bounds__(32) void k_gemm_z(const float* __restrict__ h0,
                                               const float* __restrict__ w_fc,
                                               float* __restrict__ z) {
  const int lane  = threadIdx.x;       // 0..31
  const int tile_m = blockIdx.x >> 2;  // 0..255  (rows over B*N, 16 each)
  const int tile_n = blockIdx.x & 3;   // 0..3    (cols over D, 16 each)
  const int m0   = tile_m * 16;
  const int n0   = tile_n * 16;
  const int half = lane >> 4;          // 0: lanes 0-15, 1: lanes 16-31
  const int l15  = lane & 15;

  v8f acc = {};
  for (int k0 = 0; k0 < HH; k0 += 4) {
    const float* arow = h0  + (size_t)(m0 + l15) * HH + k0 + half * 2;
    const float* brow = w_fc + (size_t)(n0 + l15) * HH + k0 + half * 2;
    v2f a; a.x = arow[0]; a.y = arow[1];
    v2f b; b.x = brow[0]; b.y = brow[1];
    acc = __builtin_amdgcn_wmma_f32_16x16x4_f32(false, a, false, b,
                                                (short)0, acc, false, false);
  }
#pragma unroll
  for (int v = 0; v < 8; ++v) {
    const int m = m0 + v + 8 * half;
    z[(size_t)m * DD + n0 + l15] = acc[v];
  }
}

// ---------------------------------------------------------------------------
// Kernel 2: sj[row] = z[row,:].a1 ; si[row] = z[row,:].a2
// a1 = w_attn[0:64], a2 = w_attn[64:128]
// ---------------------------------------------------------------------------
__global__ __launch_bounds__(256) void k_sisj(const float* __restrict__ z,
                                              const float* __restrict__ w_attn,
                                              float* __restrict__ si,
                                              float* __restrict__ sj) {
  const int row = blockIdx.x * blockDim.x + threadIdx.x;
  if (row >= BB * NN) return;
  const float* zr = z + (size_t)row * DD;
  float s1 = 0.f, s2 = 0.f;
#pragma unroll 8
  for (int d = 0; d < DD; ++d) {
    const float zv = zr[d];
    s1 += zv * w_attn[d];        // a1 -> sj
    s2 += zv * w_attn[DD + d];   // a2 -> si
  }
  sj[row] = s1;
  si[row] = s2;
}

// ---------------------------------------------------------------------------
// Kernel 3 (the bandwidth-bound one): one workgroup per (b,i) row.
// Streams e[b,i,:,:] (64 KB) once, computes se_j = e[b,i,j,:].a3,
// s = leaky_relu(si + sj[j] + se_j), and maintains an online-softmax
// running (max m, sum r) over j. Quad of lanes (t&3) covers one j's 16 floats.
// ---------------------------------------------------------------------------
__global__ __launch_bounds__(256) void k_rowpass(const float* __restrict__ e,
                                                 const float* __restrict__ w_attn,
                                                 const float* __restrict__ si,
                                                 const float* __restrict__ sj,
                                                 float* __restrict__ mrow,
                                                 float* __restrict__ rrow) {
  const int t  = threadIdx.x;
  const int bi = blockIdx.x;          // flattened b*N + i
  const int b  = bi / NN;
  const int q  = t & 3;               // position within quad
  const int jq = t >> 2;              // 0..63, j-slot per block step
  const bool leader = (q == 0);

  const float a30 = w_attn[2 * DD + q * 4 + 0];
  const float a31 = w_attn[2 * DD + q * 4 + 1];
  const float a32 = w_attn[2 * DD + q * 4 + 2];
  const float a33 = w_attn[2 * DD + q * 4 + 3];

  const float  sii  = si[bi];
  const float* sjb  = sj + (size_t)b * NN;
  const float4* erow = (const float4*)(e + (size_t)bi * NN * FEF);

  float m = -3.0e38f, r = 0.f;
  for (int jb = 0; jb < NN; jb += 64) {
    const int j = jb + jq;
    // prefetch next chunk of this row (gfx1250 global_prefetch_b8)
    if (jb + 64 < NN)
      __builtin_prefetch((const void*)&erow[(size_t)(j + 64) * 4 + q], 0, 1);

    const float4 ev = erow[(size_t)j * 4 + q];
    float p = ev.x * a30 + ev.y * a31 + ev.z * a32 + ev.w * a33;
    p += __shfl_xor(p, 1, 32);
    p += __shfl_xor(p, 2, 32);        // all 4 lanes of quad now hold se_j

    if (leader) {
      float s = sii + sjb[j] + p;
      s = s > 0.f ? s : 0.01f * s;    // leaky_relu (slope 0.01)
      const float nm = s > m ? s : m;
      r = r * __expf(m - nm) + __expf(s - nm);
      m = nm;
    }
  }

  // workgroup (m, r) merge
  __shared__ float sm[256];
  __shared__ float sr[256];
  sm[t] = m; sr[t] = r;
  __syncthreads();
  for (int off = 128; off > 0; off >>= 1) {
    if (t < off) {
      const float m2 = sm[t + off], r2 = sr[t + off];
      const float nm = m > m2 ? m : m2;
      r = r * __expf(m - nm) + r2 * __expf(m2 - nm);
      m = nm;
      sm[t] = m; sr[t] = r;
    }
    __syncthreads();
  }
  if (t == 0) { mrow[bi] = m; rrow[bi] = r; }
}

// ---------------------------------------------------------------------------
// Kernel 4: per-batch merge of (m, r) -> global M_b, Z_b
// ---------------------------------------------------------------------------
__global__ __launch_bounds__(256) void k_batchreduce(const float* __restrict__ mrow,
                                                     const float* __restrict__ rrow,
                                                     float* __restrict__ MZ) {
  const int b = blockIdx.x;
  const int t = threadIdx.x;
  float m = -3.0e38f, r = 0.f;
  for (int i = t; i < NN; i += 256) {
    const float m2 = mrow[(size_t)b * NN + i], r2 = rrow[(size_t)b * NN + i];
    const float nm = m > m2 ? m : m2;
    r = r * __expf(m - nm) + r2 * __expf(m2 - nm);
    m = nm;
  }
  __shared__ float sm[256];
  __shared__ float sr[256];
  sm[t] = m; sr[t] = r;
  __syncthreads();
  for (int off = 128; off > 0; off >>= 1) {
    if (t < off) {
      const float m2 = sm[t + off], r2 = sr[t + off];
      const float nm = m > m2 ? m : m2;
      r = r * __expf(m - nm) + r2 * __expf(m2 - nm);
      m = nm;
      sm[t] = m; sr[t] = r;
    }
    __syncthreads();
  }
  if (t == 0) { MZ[b * 2] = m; MZ[b * 2 + 1] = r; }
}

// ---------------------------------------------------------------------------
// Kernel 5: h_t[b,i,d] = (r_bi * exp(m_bi - M_b) / Z_b) * z[b,i,d]
// ---------------------------------------------------------------------------
__global__ __launch_bounds__(256) void k_out(const float* __restrict__ z,
                                             const float* __restrict__ mrow,
                                             const float* __restrict__ rrow,
                                             const float* __restrict__ MZ,
                                             float* __restrict__ out) {
  const int idx = blockIdx.x * blockDim.x + threadIdx.x;
  if (idx >= BB * NN * DD) return;
  const int bi = idx / DD;
  const int b  = bi / NN;
  const float scale = rrow[bi] * __expf(mrow[bi] - MZ[b * 2]) / MZ[b * 2 + 1];
  out[idx] = scale * z[idx];
}

// ---------------------------------------------------------------------------
// Launcher. Inputs (setup_inputs order): 0=h0 (B,N,H) f32, 1=e (B,N,N,FE) f32,
// 2=w_fc (D,H) f32, 3=w_attn (2D+FE,) f32. Output: (B,N,D) f32.
// ---------------------------------------------------------------------------
extern "C" void kernel_launch(void* const* d_in, const int* in_sizes, int n_in,
                              void* d_out, int out_size, void* d_ws, size_t ws_size,
                              hipStream_t stream) {
  const float* h0     = (const float*)d_in[0];
  const float* e      = (const float*)d_in[1];
  const float* w_fc   = (const float*)d_in[2];
  const float* w_attn = (const float*)d_in[3];
  float*       out    = (float*)d_out;

  float* ws = (float*)d_ws;
  float* z    = ws;                               // 262144 floats
  float* si   = z    + (size_t)BB * NN * DD;      // 4096
  float* sj   = si   + (size_t)BB * NN;           // 4096
  float* mrow = sj   + (size_t)BB * NN;           // 4096
  float* rrow = mrow + (size_t)BB * NN;           // 4096
  float* MZ   = rrow + (size_t)BB * NN;           // 8

  // 1) z = h0 @ w_fc^T  (WMMA f32 16x16x4), 256 row-tiles x 4 col-tiles
  k_gemm_z<<<dim3((BB * NN / 16) * (DD / 16)), dim3(32), 0, stream>>>(h0, w_fc, z);

  // 2) si/sj
  k_sisj<<<dim3((BB * NN) / 256), dim3(256), 0, stream>>>(z, w_attn, si, sj);

  // 3) stream e once, online softmax stats per row
  k_rowpass<<<dim3(BB * NN), dim3(256), 0, stream>>>(e, w_attn, si, sj, mrow, rrow);

  // 4) per-batch merge
  k_batchreduce<<<dim3(BB), dim3(256), 0, stream>>>(mrow, rrow, MZ);

  // 5) output
  k_out<<<dim3((BB * NN * DD) / 256), dim3(256), 0, stream>>>(z, mrow, rrow, MZ, out);
}